// MultiCheb_54090818126311
// MI455X (gfx1250) — compile-verified
//
#include <hip/hip_runtime.h>
#include <hip/hip_bf16.h>

// MultiCheb GraphConv forward for MI455X (gfx1250, wave32).
// All GEMMs run on the fp32 matrix pipe via V_WMMA_F32_16X16X4_F32 (exact fp32;
// problem is ~10 GFLOP / ~100MB traffic -> latency-bound at 23.3 TB/s, so fp32
// fidelity is free). Each wave register-blocks 4 M-subtiles (64x16 D tile),
// sharing the strided B fragment across 4 WMMAs. A-fragment addresses are
// derived directly from the global base pointer (no pointer array) so the
// backend emits pipelined global_load clauses instead of serialized flat loads.

#define BB 32
#define NN 128
#define CC 128
#define HEH 64

typedef float v2f __attribute__((ext_vector_type(2)));
typedef float v8f __attribute__((ext_vector_type(8)));

// ---------------------------------------------------------------------------
// WMMA fp32 GEMM: one wave computes a (16*MT) x 16 tile of D.
//   D = epilogue(alpha * A@B + beta * Csrc)
// Epilogue (in order): +bias[col], *maskRows[row], *bnScale[col]+bnShift[col], relu.
// M must be a multiple of 16*MT, Ncols of 16, K of 4 (true for every call).
// ---------------------------------------------------------------------------
template <int MT>
__global__ __launch_bounds__(32) void gemm_wmma_f32(
    const float* __restrict__ A, const float* __restrict__ B,
    const float* __restrict__ Csrc, float* __restrict__ D,
    int K, int lda, int ldb, int ldc, int ldd,
    long long sA, long long sB, long long sC, long long sD,
    float alpha, float beta,
    const float* __restrict__ bias,
    const float* __restrict__ bnScale, const float* __restrict__ bnShift,
    const float* __restrict__ maskRows, int relu)
{
  const int lane = threadIdx.x;
  const int half = lane >> 4;     // 0: lanes 0-15, 1: lanes 16-31
  const int l16  = lane & 15;
  const int tm = blockIdx.x * (16 * MT);
  const int tn = blockIdx.y * 16;
  const long long bz = blockIdx.z;

  A += bz * sA;
  B += bz * sB;
  D += bz * sD;
  if (Csrc) Csrc += bz * sC;

  // A fragment base (16x4 per subtile): lane half selects K pair {0,1} vs {2,3}
  const float* apb = A + (long long)(tm + l16) * lda + 2 * half;
  const long long arow = (long long)16 * lda;   // row stride between M-subtiles
  // B fragment (4x16), shared by all MT subtiles
  const float* bp = B + (long long)(2 * half) * ldb + tn + l16;

  v8f acc[MT];
  #pragma unroll
  for (int t = 0; t < MT; ++t) acc[t] = (v8f){};

  for (int k = 0; k < K; k += 4) {
    // Issue all fragment loads first so they can clause & pipeline.
    v2f av[MT];
    #pragma unroll
    for (int t = 0; t < MT; ++t)
      av[t] = *(const v2f*)(apb + t * arow);     // 8B-aligned (lda % 4 == 0)
    v2f bv; bv[0] = bp[0]; bv[1] = bp[ldb];
    #pragma unroll
    for (int t = 0; t < MT; ++t)
      acc[t] = __builtin_amdgcn_wmma_f32_16x16x4_f32(
          false, av[t], false, bv, (short)0, acc[t], false, false);
    apb += 4;
    bp += 4 * (long long)ldb;
  }

  const int col = tn + l16;
  const float bv2 = bias    ? bias[col]    : 0.f;
  const float sc  = bnScale ? bnScale[col] : 1.f;
  const float sh  = bnScale ? bnShift[col] : 0.f;
  #pragma unroll
  for (int t = 0; t < MT; ++t) {
    #pragma unroll
    for (int r = 0; r < 8; ++r) {
      const int row = tm + 16 * t + r + 8 * half;   // C/D layout: M = r + 8*half
      float v = alpha * acc[t][r];
      if (Csrc)     v += beta * Csrc[(long long)row * ldc + col];
      v += bv2;
      if (maskRows) v *= maskRows[row];
      if (bnScale)  v  = v * sc + sh;
      if (relu)     v  = fmaxf(v, 0.f);
      D[(long long)row * ldd + col] = v;
    }
  }
}

// ---------------------------------------------------------------------------
// Edge scores: s[b,i,j] = sum_h relu(u[b,i,h] + v[b,j,h] + eb1[h]) * eW2[h] + eb2
// grid (N, B), block 128 (one thread per j), u_i / eb1 / eW2 staged in LDS.
// ---------------------------------------------------------------------------
__global__ __launch_bounds__(128) void edge_scores(
    const float* __restrict__ U, const float* __restrict__ V,
    const float* __restrict__ eb1, const float* __restrict__ eW2,
    const float* __restrict__ eb2, float* __restrict__ S)
{
  const int i = blockIdx.x, b = blockIdx.y, j = threadIdx.x;
  __shared__ float su[HEH], sb[HEH], sw[HEH];
  if (j < HEH) {
    su[j] = U[((long long)b * NN + i) * HEH + j];
    sb[j] = eb1[j];
    sw[j] = eW2[j];
  }
  __syncthreads();
  const float* vj = V + ((long long)b * NN + j) * HEH;
  float s = eb2[0];
  #pragma unroll 8
  for (int h = 0; h < HEH; ++h)
    s += fmaxf(su[h] + vj[h] + sb[h], 0.f) * sw[h];
  S[((long long)b * NN + i) * NN + j] = s;
}

// ---------------------------------------------------------------------------
// Row-normalized predicted adjacency:
//   Apred = exp(0.5*(S+S^T)) on strict upper triangle * mask_i*mask_j
//   An = Apred / rowsum(Apred)   (rowsum==0 -> 1)
// grid (N, B), block 128; in-block tree reduction for the row sum.
// ---------------------------------------------------------------------------
__global__ __launch_bounds__(128) void edge_rownorm(
    const float* __restrict__ S, const float* __restrict__ mask,
    float* __restrict__ An)
{
  const int i = blockIdx.x, b = blockIdx.y, j = threadIdx.x;
  const float mi = mask[b * NN + i], mj = mask[b * NN + j];
  float val = 0.f;
  if (j > i) {
    const float sij = S[((long long)b * NN + i) * NN + j];
    const float sji = S[((long long)b * NN + j) * NN + i];
    val = __expf(0.5f * (sij + sji)) * mi * mj;
  }
  __shared__ float red[NN];
  red[j] = val;
  __syncthreads();
  for (int s = NN / 2; s > 0; s >>= 1) {
    if (j < s) red[j] += red[j + s];
    __syncthreads();
  }
  float rs = red[0];
  if (rs == 0.f) rs = 1.f;
  An[((long long)b * NN + i) * NN + j] = val / rs;
}

// Asym = An + An^T
__global__ __launch_bounds__(128) void symmetrize(
    const float* __restrict__ An, float* __restrict__ Asym)
{
  const int i = blockIdx.x, b = blockIdx.y, j = threadIdx.x;
  Asym[((long long)b * NN + i) * NN + j] =
      An[((long long)b * NN + i) * NN + j] +
      An[((long long)b * NN + j) * NN + i];
}

// D[b,j] = rsqrt(colsum_i M[b,i,j] + 1e-5)   (grid B, block 128)
__global__ __launch_bounds__(128) void colsum_d(
    const float* __restrict__ Mx, float* __restrict__ Dv)
{
  const int b = blockIdx.x, j = threadIdx.x;
  const float* p = Mx + (long long)b * NN * NN + j;
  float cs = 0.f;
  for (int i = 0; i < NN; ++i) cs += p[(long long)i * NN];
  Dv[b * NN + j] = rsqrtf(cs + 1e-5f);
}

// L[b,i,j] = D[b,i] * M[b,i,j] * D[b,j]
__global__ __launch_bounds__(128) void scale_lap(
    const float* __restrict__ Mx, const float* __restrict__ Dv,
    float* __restrict__ L)
{
  const int i = blockIdx.x, b = blockIdx.y, j = threadIdx.x;
  L[((long long)b * NN + i) * NN + j] =
      Dv[b * NN + i] * Mx[((long long)b * NN + i) * NN + j] * Dv[b * NN + j];
}

// BN eval fold: scale = g*rsqrt(v+eps); shift = be - m*scale
__global__ void bn_prep(const float* __restrict__ g, const float* __restrict__ be,
                        const float* __restrict__ m, const float* __restrict__ v,
                        float* __restrict__ sc, float* __restrict__ sh, int n)
{
  const int i = blockIdx.x * blockDim.x + threadIdx.x;
  if (i < n) {
    const float s = g[i] * rsqrtf(v[i] + 1e-5f);
    sc[i] = s;
    sh[i] = be[i] - m[i] * s;
  }
}

// Fold the duplicated xh row-block of W (rows [0,fin) += rows [3fin,4fin)):
// Weff rows: [xh | T1_0 | T2_0 | T1_1 | T2_1]  -> 5*fin rows
__global__ void weff_prep(const float* __restrict__ W, float* __restrict__ Weff,
                          int fin, int f)
{
  const long long idx = (long long)blockIdx.x * blockDim.x + threadIdx.x;
  const long long tot = (long long)5 * fin * f;
  if (idx >= tot) return;
  const int mrow = (int)(idx / f);
  const int c    = (int)(idx % f);
  float out;
  if (mrow < fin)
    out = W[(long long)mrow * f + c] + W[(long long)(3 * fin + mrow) * f + c];
  else if (mrow < 3 * fin)
    out = W[(long long)mrow * f + c];
  else
    out = W[(long long)(mrow + fin) * f + c];
  Weff[idx] = out;
}

// Copy [rows, fin] into column-slice 0 of Xc (leading dim ldd), float4-wide.
// fin and ldd are multiples of 4; buffers are 16B-aligned.
__global__ void copy_slice4(const float4* __restrict__ src, float4* __restrict__ dst,
                            long long total4, int fin4, int ldd4)
{
  const long long idx = (long long)blockIdx.x * blockDim.x + threadIdx.x;
  if (idx >= total4) return;
  const long long r = idx / fin4;
  const int c = (int)(idx % fin4);
  dst[r * ldd4 + c] = src[idx];
}

// pooled[b,c] = max_n H[b,n,c]   (grid B, block 512)
__global__ __launch_bounds__(512) void pool_max(const float* __restrict__ H,
                                                float* __restrict__ P)
{
  const int b = blockIdx.x, c = threadIdx.x;
  float m = -3.402823466e38f;
  for (int n = 0; n < NN; ++n)
    m = fmaxf(m, H[((long long)b * NN + n) * 512 + c]);
  P[b * 512 + c] = m;
}

// Final 32x10 FC (too small for a 16x16 WMMA tile in N): 320 dot products.
__global__ __launch_bounds__(320) void fc2_kernel(
    const float* __restrict__ Hin, const float* __restrict__ W,
    const float* __restrict__ bias, float* __restrict__ out)
{
  const int t = threadIdx.x;          // 0..319
  const int b = t / 10, o = t % 10;
  float s = bias[o];
  for (int k = 0; k < 256; ++k)
    s += Hin[b * 256 + k] * W[k * 10 + o];
  out[t] = s;
}

// ---------------------------------------------------------------------------
extern "C" void kernel_launch(void* const* d_in, const int* in_sizes, int n_in,
                              void* d_out, int out_size, void* d_ws, size_t ws_size,
                              hipStream_t stream) {
  (void)in_sizes; (void)n_in; (void)out_size; (void)ws_size;

  const float* x    = (const float*)d_in[0];   // [B,N,C]
  const float* Ain  = (const float*)d_in[1];   // [B,N,N]
  const float* mask = (const float*)d_in[2];   // [B,N]
  const float* eW1  = (const float*)d_in[3];   // [256,64]
  const float* eb1  = (const float*)d_in[4];
  const float* eW2  = (const float*)d_in[5];   // [64,1]
  const float* eb2  = (const float*)d_in[6];
  const float* gW[3] = {(const float*)d_in[7],  (const float*)d_in[13], (const float*)d_in[19]};
  const float* gb[3] = {(const float*)d_in[8],  (const float*)d_in[14], (const float*)d_in[20]};
  const float* bg[3] = {(const float*)d_in[9],  (const float*)d_in[15], (const float*)d_in[21]};
  const float* bb[3] = {(const float*)d_in[10], (const float*)d_in[16], (const float*)d_in[22]};
  const float* bm[3] = {(const float*)d_in[11], (const float*)d_in[17], (const float*)d_in[23]};
  const float* bv[3] = {(const float*)d_in[12], (const float*)d_in[18], (const float*)d_in[24]};
  const float* fW1 = (const float*)d_in[25];   // [512,256]
  const float* fb1 = (const float*)d_in[26];
  const float* fW2 = (const float*)d_in[27];   // [256,10]
  const float* fb2 = (const float*)d_in[28];
  float* out = (float*)d_out;                  // [32,10]

  // ---- workspace layout (floats) ----
  float* w = (float*)d_ws;
  size_t off = 0;
  auto take = [&](size_t n) { float* p = w + off; off += n; return p; };
  const size_t MR = (size_t)BB * NN;           // 4096 rows
  float* U    = take(MR * HEH);
  float* Vb   = take(MR * HEH);
  float* S    = take(MR * NN);
  float* An   = take(MR * NN);
  float* Asym = take(MR * NN);
  float* L0   = take(MR * NN);
  float* L1   = take(MR * NN);
  float* Dv0  = take(MR);
  float* Dv1  = take(MR);
  float* H0   = take(MR * 64);
  float* H1   = take(MR * 256);
  float* H2   = take(MR * 512);
  float* Xc   = take(MR * 1280);               // max 5*fin = 1280 (layer 2)
  float* We[3]  = {take(640 * 64), take(320 * 256), take(1280 * 512)};
  float* bnS[3] = {take(64), take(256), take(512)};
  float* bnB[3] = {take(64), take(256), take(512)};
  float* pooled = take(32 * 512);
  float* fc1o   = take(32 * 256);

  auto gemm = [&](const float* A, const float* B, const float* Cs, float* D,
                  int M, int Nc, int K, int lda, int ldb, int ldc, int ldd,
                  long long sA, long long sB, long long sC, long long sD, int batch,
                  float alpha, float beta, const float* bias,
                  const float* bns, const float* bnb, const float* mrows, int relu) {
    if (M % 64 == 0) {
      dim3 g(M / 64, Nc / 16, batch);
      gemm_wmma_f32<4><<<g, dim3(32), 0, stream>>>(A, B, Cs, D, K, lda, ldb, ldc,
                                                   ldd, sA, sB, sC, sD, alpha, beta,
                                                   bias, bns, bnb, mrows, relu);
    } else {
      dim3 g(M / 16, Nc / 16, batch);
      gemm_wmma_f32<1><<<g, dim3(32), 0, stream>>>(A, B, Cs, D, K, lda, ldb, ldc,
                                                   ldd, sA, sB, sC, sD, alpha, beta,
                                                   bias, bns, bnb, mrows, relu);
    }
  };

  // ---- parameter prep (independent of data path) ----
  const int fins[3] = {CC, 64, 256};
  const int fs[3]   = {64, 256, 512};
  for (int l = 0; l < 3; ++l) {
    bn_prep<<<(fs[l] + 255) / 256, 256, 0, stream>>>(bg[l], bb[l], bm[l], bv[l],
                                                     bnS[l], bnB[l], fs[l]);
    const long long tot = (long long)5 * fins[l] * fs[l];
    weff_prep<<<(unsigned)((tot + 255) / 256), 256, 0, stream>>>(gW[l], We[l],
                                                                 fins[l], fs[l]);
  }

  // ---- edge predictor: u = x@eW1[:C], v = x@eW1[C:] (WMMA) ----
  gemm(x, eW1,            nullptr, U,  MR, HEH, CC, CC, HEH, 0, HEH,
       0, 0, 0, 0, 1, 1.f, 0.f, nullptr, nullptr, nullptr, nullptr, 0);
  gemm(x, eW1 + CC * HEH, nullptr, Vb, MR, HEH, CC, CC, HEH, 0, HEH,
       0, 0, 0, 0, 1, 1.f, 0.f, nullptr, nullptr, nullptr, nullptr, 0);

  edge_scores<<<dim3(NN, BB), 128, 0, stream>>>(U, Vb, eb1, eW2, eb2, S);
  edge_rownorm<<<dim3(NN, BB), 128, 0, stream>>>(S, mask, An);
  symmetrize<<<dim3(NN, BB), 128, 0, stream>>>(An, Asym);

  colsum_d<<<BB, 128, 0, stream>>>(Ain, Dv0);
  colsum_d<<<BB, 128, 0, stream>>>(Asym, Dv1);
  scale_lap<<<dim3(NN, BB), 128, 0, stream>>>(Ain, Dv0, L0);
  scale_lap<<<dim3(NN, BB), 128, 0, stream>>>(Asym, Dv1, L1);

  // ---- 3 Chebyshev GraphConv layers ----
  const float* hin[3] = {x, H0, H1};
  float* hout[3] = {H0, H1, H2};
  for (int l = 0; l < 3; ++l) {
    const int fin = fins[l], f = fs[l], ld = 5 * fin;
    const long long sL = (long long)NN * NN;
    const long long sH = (long long)NN * fin;
    const long long sX = (long long)NN * ld;

    // Xc[:, :, 0:fin] = h_in   (float4-wide copy)
    const long long tot4 = (long long)MR * (fin / 4);
    copy_slice4<<<(unsigned)((tot4 + 255) / 256), 256, 0, stream>>>(
        (const float4*)hin[l], (float4*)Xc, tot4, fin / 4, ld / 4);
    // T1_0 = L0 @ h_in
    gemm(L0, hin[l], nullptr, Xc + fin, NN, fin, NN, NN, fin, 0, ld,
         sL, sH, 0, sX, BB, 1.f, 0.f, nullptr, nullptr, nullptr, nullptr, 0);
    // T2_0 = 2*L0@T1_0 - h_in
    gemm(L0, Xc + fin, hin[l], Xc + 2 * fin, NN, fin, NN, NN, ld, fin, ld,
         sL, sX, sH, sX, BB, 2.f, -1.f, nullptr, nullptr, nullptr, nullptr, 0);
    // T1_1 = L1 @ h_in
    gemm(L1, hin[l], nullptr, Xc + 3 * fin, NN, fin, NN, NN, fin, 0, ld,
         sL, sH, 0, sX, BB, 1.f, 0.f, nullptr, nullptr, nullptr, nullptr, 0);
    // T2_1 = 2*L1@T1_1 - h_in
    gemm(L1, Xc + 3 * fin, hin[l], Xc + 4 * fin, NN, fin, NN, NN, ld, fin, ld,
         sL, sX, sH, sX, BB, 2.f, -1.f, nullptr, nullptr, nullptr, nullptr, 0);
    // h_out = relu(BN(mask * (Xc @ Weff + gb)))
    gemm(Xc, We[l], nullptr, hout[l], MR, f, ld, ld, f, 0, f,
         0, 0, 0, 0, 1, 1.f, 0.f, gb[l], bnS[l], bnB[l], mask, 1);
  }

  // ---- classifier ----
  pool_max<<<BB, 512, 0, stream>>>(H2, pooled);
  gemm(pooled, fW1, nullptr, fc1o, 32, 256, 512, 512, 256, 0, 256,
       0, 0, 0, 0, 1, 1.f, 0.f, fb1, nullptr, nullptr, nullptr, 0);
  fc2_kernel<<<1, 320, 0, stream>>>(fc1o, fW2, fb2, out);
}